// CormorantCGProduct_27745488732393
// MI455X (gfx1250) — compile-verified
//
#include <hip/hip_runtime.h>

// Cormorant CG product on gfx1250.
// Roofline: output is 327 MB fp32 -> store-bound (~14us @ 23.3 TB/s).
// Compute via V_WMMA_F32_16X16X4_F32: one WMMA packs 2 CG nonzeros,
// K-slots {2h,2h+1} per lane-half h carry (+cf*ar (x) br) and
// (-cf*ai (x) bi) for the real tile (bi/br swapped for imag tile).
// CG tables built constexpr at compile time into __constant__ memory;
// term fetches are wave-uniform scalar loads + per-lane cndmask select.

typedef __attribute__((ext_vector_type(2))) float v2f;
typedef __attribute__((ext_vector_type(8))) float v8f;

#define NSAMP   1024
#define NTASK   156      // sum over triples of (2l+1)
#define MAXTERM 704      // >= selection-rule nonzeros (~478) + <=156 pad slots

struct TaskT {
  int termStart;   // into term array (even)
  int nPairs;      // WMMA steps (2 terms per step)
  int aImag;       // (2*l1+1)*16 : real->imag row offset in A block
  int bImag;       // (2*l2+1)*16 : real->imag row offset in B block
  int outOff0;     // base_l + (z*P_l + slot)*256
  int outRowStr;   // (2l+1)*P_l*256
};

struct TermT {     // 8 bytes, pair-aligned to 16B (termStart even)
  unsigned short aOff;  // absolute LDS float offset of A real row: AOFF[l1] + x*16
  unsigned short bOff;  // absolute LDS float offset of B real row: 512 + AOFF[l2] + y*16
  float coeff;          // CG coefficient (0 for pad slots)
};

struct Tables {
  TaskT task[NTASK];
  TermT term[MAXTERM];
};

// ---------- constexpr CG machinery ----------
constexpr double cfac(int x) {
  double r = 1.0;
  for (int i = 2; i <= x; ++i) r *= (double)i;
  return r;
}
constexpr double csqrt(double x) {
  double r = x > 1.0 ? x : 1.0;
  for (int i = 0; i < 48; ++i) r = 0.5 * (r + x / r);
  return r;
}
constexpr double cgc(int j1, int m1, int j2, int m2, int j, int m) {
  if (m1 + m2 != m) return 0.0;
  int dj = j1 > j2 ? j1 - j2 : j2 - j1;
  if (j < dj || j > j1 + j2) return 0.0;
  double pref = csqrt((2.0 * j + 1.0)
      * cfac(j + j1 - j2) * cfac(j - j1 + j2) * cfac(j1 + j2 - j) / cfac(j1 + j2 + j + 1)
      * cfac(j + m) * cfac(j - m) * cfac(j1 - m1) * cfac(j1 + m1)
      * cfac(j2 - m2) * cfac(j2 + m2));
  int kmin = 0;
  if (j2 - j - m1 > kmin) kmin = j2 - j - m1;
  if (j1 - j + m2 > kmin) kmin = j1 - j + m2;
  int kmax = j1 + j2 - j;
  if (j1 - m1 < kmax) kmax = j1 - m1;
  if (j2 + m2 < kmax) kmax = j2 + m2;
  double s = 0.0;
  for (int k = kmin; k <= kmax; ++k) {
    double d = cfac(k) * cfac(j1 + j2 - j - k) * cfac(j1 - m1 - k) * cfac(j2 + m2 - k)
             * cfac(j - j2 + m1 + k) * cfac(j - j1 - m2 + k);
    s += ((k & 1) ? -1.0 : 1.0) / d;
  }
  return pref * s;
}

constexpr Tables buildTables() {
  Tables T{};
  constexpr int AOFFc[4] = {0, 32, 128, 288};          // per-l LDS block starts (A region)
  constexpr int PNUM[4]  = {4, 9, 11, 10};             // #(l1,l2) pairs feeding degree l
  constexpr int OBASE[4] = {0, 2097152, 16252928, 45088768}; // flat out base per degree
  int slot[4] = {0, 0, 0, 0};
  int term = 0, tix = 0;
  for (int l1 = 0; l1 <= 3; ++l1)
    for (int l2 = 0; l2 <= 3; ++l2) {
      int lmin = l1 > l2 ? l1 - l2 : l2 - l1;
      int lmax = (l1 + l2 < 3) ? (l1 + l2) : 3;
      for (int l = lmin; l <= lmax; ++l) {
        for (int z = 0; z < 2 * l + 1; ++z) {
          int start = term, cnt = 0;
          for (int x = 0; x < 2 * l1 + 1; ++x)
            for (int y = 0; y < 2 * l2 + 1; ++y) {
              int m1 = x - l1, m2 = y - l2, m = z - l;
              if (m1 + m2 == m) {
                double c = cgc(l1, m1, l2, m2, l, m);
                double ac = c < 0 ? -c : c;
                if (ac > 1e-10) {
                  T.term[term].aOff  = (unsigned short)(AOFFc[l1] + x * 16);
                  T.term[term].bOff  = (unsigned short)(512 + AOFFc[l2] + y * 16);
                  T.term[term].coeff = (float)c;
                  ++term; ++cnt;
                }
              }
            }
          if (cnt & 1) {  // pad to even so lane-half h picks term 2p+h
            T.term[term].aOff = 0; T.term[term].bOff = 512; T.term[term].coeff = 0.0f;
            ++term; ++cnt;
          }
          T.task[tix].termStart = start;
          T.task[tix].nPairs    = cnt / 2;
          T.task[tix].aImag     = (2 * l1 + 1) * 16;
          T.task[tix].bImag     = (2 * l2 + 1) * 16;
          T.task[tix].outOff0   = OBASE[l] + (z * PNUM[l] + slot[l]) * 256;
          T.task[tix].outRowStr = (2 * l + 1) * PNUM[l] * 256;
          ++tix;
        }
        ++slot[l];
      }
    }
  return T;
}

static __constant__ Tables gTab = buildTables();

// ---------- kernel ----------
__global__ __launch_bounds__(256) void cgprod_wmma_kernel(
    const float* __restrict__ A0, const float* __restrict__ A1,
    const float* __restrict__ A2, const float* __restrict__ A3,
    const float* __restrict__ B0, const float* __restrict__ B1,
    const float* __restrict__ B2, const float* __restrict__ B3,
    float* __restrict__ out)
{
  __shared__ float smem[1024];  // [A rows 0..511][B rows 512..1023], 4KB
  const int n = blockIdx.x;

  const float* Ap[4] = {A0, A1, A2, A3};
  const float* Bp[4] = {B0, B1, B2, B3};
  const int AOFFk[4] = {0, 32, 128, 288};

  // Stage this sample's A/B rows: src layout (reim, N, 2l+1, 16).
  for (int l = 0; l < 4; ++l) {
    const int dim16 = (2 * l + 1) * 16;
    const int cnt = 2 * dim16;
    const float* as = Ap[l] + n * dim16;
    const float* bs = Bp[l] + n * dim16;
    for (int e = threadIdx.x; e < cnt; e += 256) {
      const int reim = (e >= dim16) ? 1 : 0;
      const int r = e - reim * dim16;
      const int src = reim * NSAMP * dim16 + r;
      smem[AOFFk[l] + e]       = as[src];
      smem[512 + AOFFk[l] + e] = bs[src];
    }
  }
  __syncthreads();

  const int wave = threadIdx.x >> 5;
  const int lane = threadIdx.x & 31;
  const int col  = lane & 15;      // tau column (A-side c for loads, B-side d)
  const int h    = lane >> 4;      // lane half: owns K-slots {2h, 2h+1}

  for (int t = wave; t < NTASK; t += 8) {   // wave-uniform -> EXEC stays all-ones
    const TaskT tk = gTab.task[t];

    v8f accR = {0.f, 0.f, 0.f, 0.f, 0.f, 0.f, 0.f, 0.f};
    v8f accI = {0.f, 0.f, 0.f, 0.f, 0.f, 0.f, 0.f, 0.f};

    const TermT* tp = &gTab.term[tk.termStart];
    for (int p = 0; p < tk.nPairs; ++p) {
      // Wave-uniform scalar loads of the term pair; per-lane select by half.
      const TermT t0 = tp[2 * p];
      const TermT t1 = tp[2 * p + 1];
      const int   aOff = h ? (int)t1.aOff : (int)t0.aOff;
      const int   bOff = h ? (int)t1.bOff : (int)t0.bOff;
      const float cf   = h ? t1.coeff     : t0.coeff;

      const float ar = smem[aOff + col];
      const float ai = smem[aOff + tk.aImag + col];
      const float br = smem[bOff + col];
      const float bi = smem[bOff + tk.bImag + col];

      const float cfar = cf * ar;
      const float cfai = cf * ai;

      // Real tile: +cf*ar(x)br  (K=2h)   -cf*ai(x)bi  (K=2h+1)
      v2f Amr = {cfar, -cfai};
      v2f Bmr = {br, bi};
      // Imag tile: +cf*ar(x)bi  (K=2h)   +cf*ai(x)br  (K=2h+1)
      v2f Ami = {cfar, cfai};
      v2f Bmi = {bi, br};

      accR = __builtin_amdgcn_wmma_f32_16x16x4_f32(
          false, Amr, false, Bmr, (short)0, accR, false, false);
      accI = __builtin_amdgcn_wmma_f32_16x16x4_f32(
          false, Ami, false, Bmi, (short)0, accI, false, false);
    }

    // D layout: VGPR i -> row (h*8 + i), col = lane&15. Streaming NT stores.
    const int baseR = tk.outOff0 + n * tk.outRowStr;
    const int baseI = tk.outOff0 + (NSAMP + n) * tk.outRowStr;
    const int rb = h * 128;  // (h*8)*16
#pragma unroll
    for (int i = 0; i < 8; ++i) {
      __builtin_nontemporal_store(accR[i], &out[baseR + rb + i * 16 + col]);
      __builtin_nontemporal_store(accI[i], &out[baseI + rb + i * 16 + col]);
    }
  }
}

extern "C" void kernel_launch(void* const* d_in, const int* in_sizes, int n_in,
                              void* d_out, int out_size, void* d_ws, size_t ws_size,
                              hipStream_t stream) {
  // setup_inputs() dict order: A0, B0, A1, B1, A2, B2, A3, B3
  const float* A0 = (const float*)d_in[0];
  const float* B0 = (const float*)d_in[1];
  const float* A1 = (const float*)d_in[2];
  const float* B1 = (const float*)d_in[3];
  const float* A2 = (const float*)d_in[4];
  const float* B2 = (const float*)d_in[5];
  const float* A3 = (const float*)d_in[6];
  const float* B3 = (const float*)d_in[7];
  float* out = (float*)d_out;
  (void)in_sizes; (void)n_in; (void)out_size; (void)d_ws; (void)ws_size;

  cgprod_wmma_kernel<<<dim3(NSAMP), dim3(256), 0, stream>>>(
      A0, A1, A2, A3, B0, B1, B2, B3, out);
}